// GlobalHierarchyCPCCLoss_37074157699118
// MI455X (gfx1250) — compile-verified
//
#include <hip/hip_runtime.h>
#include <hip/hip_bf16.h>

#define GN   1024      // groups
#define MN   128       // members per group (max)
#define DN   128       // embedding dim
#define EPSF 1e-5f

typedef float v2f __attribute__((ext_vector_type(2)));
typedef float v8f __attribute__((ext_vector_type(8)));

// ---------------------------------------------------------------------------
// Kernel 0: zero the 5 Pearson accumulators
// ---------------------------------------------------------------------------
__global__ void init_acc_kernel(float* __restrict__ acc) {
    if (threadIdx.x < 8) acc[threadIdx.x] = 0.0f;
}

// ---------------------------------------------------------------------------
// Kernel 1: per-group Einstein-midpoint mean in the Poincare ball.
// 1 block per group, 128 threads (4 waves). Each wave processes 32 members;
// a whole wave handles one member row: lane l loads float4 of dims [4l,4l+4).
// ---------------------------------------------------------------------------
__global__ void reps_kernel(const float* __restrict__ emb,
                            const int*   __restrict__ midx,
                            const unsigned char* __restrict__ mmask,  // numpy bool layout
                            float* __restrict__ reps,
                            float* __restrict__ sqn) {
    const int g    = blockIdx.x;
    const int tid  = threadIdx.x;      // 0..127
    const int wave = tid >> 5;         // 0..3
    const int lane = tid & 31;

    __shared__ float accLDS[4 * DN];
    __shared__ float wsumLDS[4];
    __shared__ float red[DN];

    const float maxn = 1.0f - EPSF;

    float4 acc = make_float4(0.f, 0.f, 0.f, 0.f);
    float  wsum = 0.f;

    // each wave: members [wave*32, wave*32+32)
    for (int m = wave * 32; m < wave * 32 + 32; ++m) {
        if (!mmask[g * MN + m]) continue;               // wave-uniform branch
        const int idx = midx[g * MN + m];
        const float4 p = ((const float4*)(emb + (long long)idx * DN))[lane];

        // raw squared norm (wave32 reduction)
        float nrm2 = p.x*p.x + p.y*p.y + p.z*p.z + p.w*p.w;
        #pragma unroll
        for (int s = 16; s > 0; s >>= 1) nrm2 += __shfl_xor(nrm2, s, 32);

        // project_to_ball on the member
        const float nrm   = sqrtf(nrm2);
        const float scale = (nrm > maxn) ? (maxn / fmaxf(nrm, EPSF)) : 1.0f;
        const float sq    = nrm2 * scale * scale;       // |p_proj|^2

        // Poincare -> Klein; Lorentz factor (all scalar, no 2nd reduction)
        const float t     = 2.0f / (1.0f + sq);
        const float k_sq  = t * t * sq;
        const float gamma = rsqrtf(fmaxf(1.0f - k_sq, EPSF));
        const float c     = gamma * t * scale;          // k = (t*scale) * p_raw

        acc.x += c * p.x;  acc.y += c * p.y;
        acc.z += c * p.z;  acc.w += c * p.w;
        wsum  += gamma;
    }

    accLDS[wave * DN + 4 * lane + 0] = acc.x;
    accLDS[wave * DN + 4 * lane + 1] = acc.y;
    accLDS[wave * DN + 4 * lane + 2] = acc.z;
    accLDS[wave * DN + 4 * lane + 3] = acc.w;
    if (lane == 0) wsumLDS[wave] = wsum;
    __syncthreads();

    // combine 4 wave-accumulators: thread tid owns dim tid
    const float s = accLDS[tid] + accLDS[DN + tid] + accLDS[2 * DN + tid] + accLDS[3 * DN + tid];
    const float W = wsumLDS[0] + wsumLDS[1] + wsumLDS[2] + wsumLDS[3];
    const float mean_k = s / fmaxf(W, EPSF);

    // block reduction of |mean_k|^2
    red[tid] = mean_k * mean_k;
    __syncthreads();
    #pragma unroll
    for (int st = 64; st > 0; st >>= 1) {
        if (tid < st) red[tid] += red[tid + st];
        __syncthreads();
    }
    const float mk_sq = red[0];

    // Klein -> Poincare, then project_to_ball (all scalar except final scale)
    const float ds    = 1.0f + sqrtf(fmaxf(1.0f - mk_sq, EPSF));
    const float mp    = mean_k / ds;
    const float mp_sq = mk_sq / (ds * ds);
    const float nrm   = sqrtf(mp_sq);
    const float sc    = (nrm > maxn) ? (maxn / fmaxf(nrm, EPSF)) : 1.0f;

    reps[g * DN + tid] = mp * sc;
    if (tid == 0) sqn[g] = mp_sq * sc * sc;
}

// ---------------------------------------------------------------------------
// Kernel 2: upper-triangular 16x16 tiles of reps @ reps^T via
// V_WMMA_F32_16X16X4_F32, fused hyperbolic distance + Pearson partial sums.
// One wave per tile; 2080 tiles total (64*65/2), grid = 260 blocks x 8 waves.
// ---------------------------------------------------------------------------
__global__ void dist_corr_kernel(const float* __restrict__ reps,
                                 const float* __restrict__ sqn,
                                 const float* __restrict__ target,
                                 float* __restrict__ acc) {
    const int wave = threadIdx.x >> 5;
    const int lane = threadIdx.x & 31;
    const int TG   = GN / 16;                       // 64 tile rows
    const int NT   = TG * (TG + 1) / 2;             // 2080 tiles
    const int t    = blockIdx.x * (blockDim.x >> 5) + wave;
    if (t >= NT) return;                            // wave-uniform

    // decode (ti, tj) with ti <= tj from linear upper-tri index (SALU loop)
    int ti = 0, rem = t;
    while (rem >= TG - ti) { rem -= TG - ti; ++ti; }
    const int tj = ti + rem;

    // WMMA f32 16x16x4 operand layout (ISA 7.12.2):
    //   lane = half*16 + idx ; fragment.xy = row[idx], K = k0 + 2*half + {0,1}
    const int idx16 = lane & 15;
    const int koff  = (lane >> 4) * 2;
    const float* Arow = reps + (long long)(ti * 16 + idx16) * DN + koff;
    const float* Brow = reps + (long long)(tj * 16 + idx16) * DN + koff;

    v8f c = {0.f, 0.f, 0.f, 0.f, 0.f, 0.f, 0.f, 0.f};
    #pragma unroll
    for (int k0 = 0; k0 < DN; k0 += 4) {
        const v2f a = *(const v2f*)(Arow + k0);
        const v2f b = *(const v2f*)(Brow + k0);
        c = __builtin_amdgcn_wmma_f32_16x16x4_f32(
                /*neg_a=*/false, a, /*neg_b=*/false, b,
                /*c_mod=*/(short)0, c, /*reuse_a=*/false, /*reuse_b=*/false);
    }

    // C layout: VGPR r, lane l -> M = r + 8*(l>>4), N = l&15
    const int half = lane >> 4;
    const int n    = lane & 15;
    const int j    = tj * 16 + n;
    const float sqj = sqn[j];

    float Sx = 0.f, Sy = 0.f, Sxx = 0.f, Syy = 0.f, Sxy = 0.f;
    #pragma unroll
    for (int r = 0; r < 8; ++r) {
        const int m = r + half * 8;
        const int i = ti * 16 + m;
        if (j > i) {                                // strict upper triangle
            const float dot = c[r];
            const float sqi = sqn[i];
            const float d2  = fmaxf(sqi + sqj - 2.0f * dot, 0.0f);
            const float den = fmaxf((1.0f - sqi) * (1.0f - sqj), EPSF);
            const float arg = fmaxf(1.0f + 2.0f * d2 / den, 1.0f + 1e-7f);
            const float dst = logf(arg + sqrtf(arg * arg - 1.0f));   // arccosh
            const int cidx  = i * GN - (i * (i + 1)) / 2 + (j - i - 1);
            const float y   = target[cidx];
            Sx += dst;  Sy += y;
            Sxx += dst * dst;  Syy += y * y;  Sxy += dst * y;
        }
    }

    #pragma unroll
    for (int s = 16; s > 0; s >>= 1) {
        Sx  += __shfl_xor(Sx,  s, 32);
        Sy  += __shfl_xor(Sy,  s, 32);
        Sxx += __shfl_xor(Sxx, s, 32);
        Syy += __shfl_xor(Syy, s, 32);
        Sxy += __shfl_xor(Sxy, s, 32);
    }
    if (lane == 0) {
        atomicAdd(&acc[0], Sx);
        atomicAdd(&acc[1], Sy);
        atomicAdd(&acc[2], Sxx);
        atomicAdd(&acc[3], Syy);
        atomicAdd(&acc[4], Sxy);
    }
}

// ---------------------------------------------------------------------------
// Kernel 3: finalize 1 - Pearson correlation
// ---------------------------------------------------------------------------
__global__ void final_kernel(const float* __restrict__ acc, float* __restrict__ out) {
    const float P   = (float)((long long)GN * (GN - 1) / 2);
    const float Sx  = acc[0], Sy = acc[1], Sxx = acc[2], Syy = acc[3], Sxy = acc[4];
    const float num = Sxy - Sx * Sy / P;
    const float den = sqrtf((Sxx - Sx * Sx / P) * (Syy - Sy * Sy / P) + 1e-12f);
    out[0] = 1.0f - num / den;
}

// ---------------------------------------------------------------------------
extern "C" void kernel_launch(void* const* d_in, const int* in_sizes, int n_in,
                              void* d_out, int out_size, void* d_ws, size_t ws_size,
                              hipStream_t stream) {
    const float*         emb    = (const float*)d_in[0];            // [N, D]
    const float*         target = (const float*)d_in[1];            // [P]
    const int*           midx   = (const int*)d_in[2];              // [G, M]
    const unsigned char* mmask  = (const unsigned char*)d_in[3];    // [G, M] bool

    float* reps = (float*)d_ws;                 // G*D floats (512 KB)
    float* sqn  = reps + (size_t)GN * DN;       // G floats
    float* acc  = sqn + GN;                     // 5 floats (+pad)

    init_acc_kernel<<<1, 32, 0, stream>>>(acc);
    reps_kernel<<<GN, 128, 0, stream>>>(emb, midx, mmask, reps, sqn);

    const int NT = (GN / 16) * (GN / 16 + 1) / 2;   // 2080 tiles
    dist_corr_kernel<<<(NT + 7) / 8, 256, 0, stream>>>(reps, sqn, target, acc);

    final_kernel<<<1, 1, 0, stream>>>(acc, (float*)d_out);
}